// MoELayer_41738492183144
// MI455X (gfx1250) — compile-verified
//
#include <hip/hip_runtime.h>
#include <math.h>

typedef float v2f __attribute__((ext_vector_type(2)));
typedef float v8f __attribute__((ext_vector_type(8)));

#define D_DIM   2048
#define FF_DIM  5632
#define T_TOK   16384   // B*S = 4*4096
#define TM      32      // tokens per tile (2 M-subtiles of 16)
#define NWAVES  16
#define BLOCK   (NWAVES * 32)
#define FC      (NWAVES * 16)   // FF chunk = 256 cols; 5632 = 22*256
#define XS_STRIDE 2052  // 2048 + 4 pad (banks)
#define HS_STRIDE 260   // 256 + 4 pad

__launch_bounds__(BLOCK, 1)
__global__ void moe_swiglu_wmma(const float* __restrict__ x,
                                const int*   __restrict__ mask,
                                const float* __restrict__ wg,
                                const float* __restrict__ wu,
                                const float* __restrict__ wd,
                                float*       __restrict__ out)
{
    __shared__ float xs[TM * XS_STRIDE]; // x tile [32][2052]  (~257 KB)
    __shared__ float hs[TM * HS_STRIDE]; // h tile [32][260]   (~32.5 KB)

    const int tile = blockIdx.x;
    const int e    = blockIdx.y;      // expert id (0 or 1)
    const int t0   = tile * TM;

    // Uniform early-exit: skip tiles with no tokens routed to this expert.
    int need = 0;
    for (int i = 0; i < TM; ++i) need |= (mask[t0 + i] == e) ? 1 : 0;
    if (!need) return;

    const float* wgE = wg + (size_t)e * D_DIM * FF_DIM;
    const float* wuE = wu + (size_t)e * D_DIM * FF_DIM;
    const float* wdE = wd + (size_t)e * FF_DIM * D_DIM;

    const int tid  = threadIdx.x;
    const int wave = tid >> 5;
    const int lane = tid & 31;
    const int l16  = lane & 15;
    const int hi   = lane >> 4;       // half-wave: 0 -> K+{0,1}, 1 -> K+{2,3}

    // ---- Stage x[32][2048] into LDS (float4 vector copies) ----
    {
        const float4* xg = (const float4*)(x + (size_t)t0 * D_DIM);
        for (int i = tid; i < TM * D_DIM / 4; i += BLOCK) {
            float4 v = xg[i];
            int row = (i * 4) / D_DIM;
            int col = (i * 4) % D_DIM;
            *(float4*)&xs[row * XS_STRIDE + col] = v;
        }
    }
    __syncthreads();

    // Output accumulators: wave owns out cols [wave*128, +127] for 32 rows.
    // 16 waves x 128 cols = full D = 2048 columns covered.
    v8f acc0[8], acc1[8];
    #pragma unroll
    for (int j = 0; j < 8; ++j) { acc0[j] = (v8f){}; acc1[j] = (v8f){}; }

    const int hcol = wave * 16 + l16;   // h column this lane produces (0..255)
    const int n0   = wave * 128;        // output column base for this wave

    for (int ffc = 0; ffc < FF_DIM; ffc += FC) {
        // ===== gate & up GEMMs: [32 tok] x [16 ff cols per wave], K = D =====
        v8f ag0 = (v8f){}, au0 = (v8f){};   // rows 0..15
        v8f ag1 = (v8f){}, au1 = (v8f){};   // rows 16..31
        const float* wgp   = wgE + (size_t)(ffc + hcol);
        const float* wup   = wuE + (size_t)(ffc + hcol);
        const float* xrow0 = &xs[l16 * XS_STRIDE + 2 * hi];
        const float* xrow1 = &xs[(16 + l16) * XS_STRIDE + 2 * hi];
        #pragma unroll 4
        for (int k = 0; k < D_DIM; k += 4) {
            v2f a0 = *(const v2f*)(xrow0 + k);   // A tile rows 0..15
            v2f a1 = *(const v2f*)(xrow1 + k);   // A tile rows 16..31
            int kr = k + 2 * hi;
            v2f bg, bu;                          // B fragments shared by both M-tiles
            bg.x = wgp[(size_t)kr * FF_DIM];
            bg.y = wgp[(size_t)(kr + 1) * FF_DIM];
            bu.x = wup[(size_t)kr * FF_DIM];
            bu.y = wup[(size_t)(kr + 1) * FF_DIM];
            ag0 = __builtin_amdgcn_wmma_f32_16x16x4_f32(false, a0, false, bg, (short)0, ag0, false, false);
            ag1 = __builtin_amdgcn_wmma_f32_16x16x4_f32(false, a1, false, bg, (short)0, ag1, false, false);
            au0 = __builtin_amdgcn_wmma_f32_16x16x4_f32(false, a0, false, bu, (short)0, au0, false, false);
            au1 = __builtin_amdgcn_wmma_f32_16x16x4_f32(false, a1, false, bu, (short)0, au1, false, false);
        }

        // silu(gate) * up, elementwise (v_exp + v_rcp, dual-issue friendly)
        v8f h0, h1;
        #pragma unroll
        for (int r = 0; r < 8; ++r) {
            float g0 = ag0[r];
            float g1 = ag1[r];
            h0[r] = g0 * __builtin_amdgcn_rcpf(1.0f + __expf(-g0)) * au0[r];
            h1[r] = g1 * __builtin_amdgcn_rcpf(1.0f + __expf(-g1)) * au1[r];
        }

        __syncthreads();   // all waves done reading hs from previous chunk
        // C-layout -> LDS: lane holds col hcol, rows r + 8*hi (and +16)
        #pragma unroll
        for (int r = 0; r < 8; ++r) {
            int m = r + 8 * hi;
            hs[m * HS_STRIDE + hcol]        = h0[r];
            hs[(16 + m) * HS_STRIDE + hcol] = h1[r];
        }
        __syncthreads();   // hs chunk fully written

        // ===== down GEMM accumulate: out[32, n0..n0+127] += h[32,256] @ wd =====
        const float* wdp   = wdE + (size_t)ffc * D_DIM + n0 + l16;
        const float* hrow0 = &hs[l16 * HS_STRIDE + 2 * hi];
        const float* hrow1 = &hs[(16 + l16) * HS_STRIDE + 2 * hi];
        #pragma unroll 2
        for (int k2 = 0; k2 < FC; k2 += 4) {
            v2f a0 = *(const v2f*)(hrow0 + k2);  // A from LDS h tile
            v2f a1 = *(const v2f*)(hrow1 + k2);
            int kr = k2 + 2 * hi;
            const float* b0 = wdp + (size_t)kr * D_DIM;
            const float* b1 = b0 + D_DIM;
            #pragma unroll
            for (int j = 0; j < 8; ++j) {
                v2f b;                            // B fragment shared by both M-tiles
                b.x = b0[j * 16];
                b.y = b1[j * 16];
                acc0[j] = __builtin_amdgcn_wmma_f32_16x16x4_f32(false, a0, false, b, (short)0, acc0[j], false, false);
                acc1[j] = __builtin_amdgcn_wmma_f32_16x16x4_f32(false, a1, false, b, (short)0, acc1[j], false, false);
            }
        }
    }

    // ===== predicated store: only rows routed to this expert =====
    #pragma unroll
    for (int r = 0; r < 8; ++r) {
        int m = r + 8 * hi;
        int token = t0 + m;
        if (mask[token] == e) {
            float* orow = out + (size_t)token * D_DIM + n0 + l16;
            #pragma unroll
            for (int j = 0; j < 8; ++j)
                orow[j * 16] = acc0[j][r];
        }
        int token1 = t0 + 16 + m;
        if (mask[token1] == e) {
            float* orow = out + (size_t)token1 * D_DIM + n0 + l16;
            #pragma unroll
            for (int j = 0; j < 8; ++j)
                orow[j * 16] = acc1[j][r];
        }
    }
}

extern "C" void kernel_launch(void* const* d_in, const int* in_sizes, int n_in,
                              void* d_out, int out_size, void* d_ws, size_t ws_size,
                              hipStream_t stream) {
    const float* x    = (const float*)d_in[0];  // hidden_states [B,S,D]
    const int*   mask = (const int*)  d_in[1];  // routing_mask  [B,S]
    const float* wg   = (const float*)d_in[2];  // w_gate [E,D,FF]
    const float* wu   = (const float*)d_in[3];  // w_up   [E,D,FF]
    const float* wd   = (const float*)d_in[4];  // w_down [E,FF,D]
    float* out = (float*)d_out;

    dim3 grid(T_TOK / TM, 2);   // 512 token tiles x 2 experts
    moe_swiglu_wmma<<<grid, BLOCK, 0, stream>>>(x, mask, wg, wu, wd, out);
}